// EnVAE_68487548502600
// MI455X (gfx1250) — compile-verified
//
#include <hip/hip_runtime.h>
#include <hip/hip_bf16.h>

// ---------------------------------------------------------------------------
// EnVAE forward on gfx1250 (MI455X), fp32 WMMA path (V_WMMA_F32_16X16X4_F32).
// Memory-bound (~300MB HBM traffic, ~14 GFLOP) -> fp32 exactness is free.
// This revision software-pipelines the fragment loads so the wave issues all
// 2+2*NT loads of step k+1 before consuming step k's fragments (no more
// s_wait_loadcnt 0 before every wmma), and makes every stride constexpr so
// address math folds into immediate offsets.
// ---------------------------------------------------------------------------

typedef __attribute__((ext_vector_type(2))) float v2f;
typedef __attribute__((ext_vector_type(8))) float v8f;

// Grouped GEMM: C[m, g, n] = act( sum_k A[m, g, k] * B[g, k, n] + bias[g, n] )
// One wave computes a 16 x (16*NT) tile. Strides are template params so the
// same kernel does the Xg gather (LDA_K=4) and the recon scatter (LDC_COL=4).
//
// WMMA f32 16x16x4 lane layout (ISA 7.12.2):
//   A (16x4): lane&15 = M;  lanes 0-15 hold K=0,1  lanes 16-31 hold K=2,3
//   B (4x16): lane&15 = N;  same K split
//   C/D (16x16): v8f, VGPR v -> M = v + 8*(lane>>4), N = lane&15
template <int NT, int LDA_ROW, int LDA_G, int LDA_K,
          int LDB_G, int LDB_K, int LDBIAS_G,
          int LDC_ROW, int LDC_G, int LDC_COL, int RELU>
__global__ __launch_bounds__(32) void wmma_gemm_grouped(
    const float* __restrict__ A, const float* __restrict__ Bm,
    const float* __restrict__ bias, float* __restrict__ C, int K)
{
    const int lane  = threadIdx.x;          // 0..31, one wave per block
    const int M0    = blockIdx.x * 16;
    const int N0    = blockIdx.y * (16 * NT);
    const int g     = blockIdx.z;
    const int mn    = lane & 15;            // M for A, N for B/C
    const int khalf = (lane >> 4) << 1;     // 0 (lanes 0-15) or 2 (lanes 16-31)

    v8f acc[NT] = {};

    const float* Ab = A  + (long)(M0 + mn) * LDA_ROW + (long)g * LDA_G
                         + (long)khalf * LDA_K;
    const float* Bb = Bm + (long)g * LDB_G + (long)khalf * LDB_K + N0 + mn;

    auto loadA = [&](const float* p) -> v2f {
        if constexpr (LDA_K == 1) {
            return *(const v2f*)p;           // adjacent + 8B aligned -> b64 load
        } else {
            v2f r; r.x = p[0]; r.y = p[LDA_K]; return r;
        }
    };

    // ---- software pipeline: prologue loads for k = 0 ----
    v2f a = loadA(Ab);
    v2f bf[NT];
#pragma unroll
    for (int t = 0; t < NT; ++t) {
        bf[t].x = Bb[t * 16];
        bf[t].y = Bb[t * 16 + LDB_K];
    }

    for (int k = 0; k < K - 4; k += 4) {
        Ab += 4 * LDA_K;
        Bb += 4 * LDB_K;
        // issue ALL of next step's loads first...
        v2f an = loadA(Ab);
        v2f bn[NT];
#pragma unroll
        for (int t = 0; t < NT; ++t) {
            bn[t].x = Bb[t * 16];
            bn[t].y = Bb[t * 16 + LDB_K];
        }
        // ...then consume the current fragments while next loads are in flight
#pragma unroll
        for (int t = 0; t < NT; ++t)
            acc[t] = __builtin_amdgcn_wmma_f32_16x16x4_f32(
                false, a, false, bf[t], (short)0, acc[t], false, false);
        a = an;
#pragma unroll
        for (int t = 0; t < NT; ++t) bf[t] = bn[t];
    }
    // pipeline drain: last k-step
#pragma unroll
    for (int t = 0; t < NT; ++t)
        acc[t] = __builtin_amdgcn_wmma_f32_16x16x4_f32(
            false, a, false, bf[t], (short)0, acc[t], false, false);

    // ---- epilogue: bias (+ ReLU) and strided/scattered store ----
    const int mb = M0 + ((lane >> 4) << 3);
#pragma unroll
    for (int t = 0; t < NT; ++t) {
        float bv = bias[(long)g * LDBIAS_G + N0 + t * 16 + mn];
#pragma unroll
        for (int v = 0; v < 8; ++v) {
            float r = acc[t][v] + bv;
            if (RELU) r = fmaxf(r, 0.0f);
            C[(long)(mb + v) * LDC_ROW + (long)g * LDC_G
              + (long)(N0 + t * 16 + mn) * LDC_COL] = r;
        }
    }
}

// Posterior combine: only the SELECTED subset per batch element is needed.
//   tau  = 1 + sum_{g in S} exp(-logvar_g)
//   mean = (sum_{g in S} exp(-logvar_g) * mean_g) / tau
//   z    = mean + eps * exp(0.5 * (-log tau)) = mean + eps * rsqrt(tau)
// Powerset order (r=1..4 combinations) packed as 4-bit masks:
//   {1,2,4,8, 3,5,9,6,10,12, 7,11,13,14, 15}
__global__ void envae_combine(const float* __restrict__ ml,
                              const float* __restrict__ eps,
                              const int* __restrict__ subset_idx,
                              float* __restrict__ z)
{
    const unsigned long long PACK = 0x0FEDB7CA69538421ULL;
    int idx = blockIdx.x * blockDim.x + threadIdx.x;   // B*LAT = 524288
    int b = idx >> 6;
    int l = idx & 63;
    int s = subset_idx[b];
    unsigned mask = (unsigned)((PACK >> (4 * s)) & 0xFull);

    float tau = 1.0f, num = 0.0f;
#pragma unroll
    for (int g = 0; g < 4; ++g) {
        if (mask & (1u << g)) {
            float mean = ml[(long)b * 512 + g * 128 + l];
            float lv   = ml[(long)b * 512 + g * 128 + 64 + l];
            float t = expf(-lv);
            tau += t;
            num = fmaf(t, mean, num);
        }
    }
    z[idx] = num / tau + eps[idx] * rsqrtf(tau);
}

extern "C" void kernel_launch(void* const* d_in, const int* in_sizes, int n_in,
                              void* d_out, int out_size, void* d_ws, size_t ws_size,
                              hipStream_t stream) {
    const float* X    = (const float*)d_in[0];   // (8192, 4096)
    const float* eps  = (const float*)d_in[1];   // (8192, 64)
    const int*   sidx = (const int*)  d_in[2];   // (8192,)
    const float* ew1  = (const float*)d_in[3];   // (4, 1024, 128)
    const float* eb1  = (const float*)d_in[4];   // (4, 128)
    const float* ew2  = (const float*)d_in[5];   // (4, 128, 128)
    const float* eb2  = (const float*)d_in[6];   // (4, 128)
    const float* dw1  = (const float*)d_in[7];   // (4, 64, 64)
    const float* db1  = (const float*)d_in[8];   // (4, 64)
    const float* dw2  = (const float*)d_in[9];   // (4, 64, 1024)
    const float* db2  = (const float*)d_in[10];  // (4, 1024)
    float* recon = (float*)d_out;                // (8192, 4096)

    char* ws = (char*)d_ws;
    float* Hbuf = (float*)(ws);                          // (8192,4,128) 16 MB
    float* mlb  = (float*)(ws + ((size_t)16 << 20));     // (8192,4,128) 16 MB
    float* hd   = (float*)(ws);                          // (8192,4,64)   8 MB (reuses Hbuf)
    float* z    = (float*)(ws + ((size_t)8  << 20));     // (8192,64)     2 MB (reuses Hbuf)

    // enc1: h = relu(Xg @ enc_w1 + b1);  Xg[b,g,i] = X[b*4096 + g + 4*i]
    //   M=8192, N=128 (NT=8 -> grid.y=1), K=1024
    wmma_gemm_grouped<8, 4096, 1, 4, 1024 * 128, 128, 128, 512, 128, 1, 1>
        <<<dim3(512, 1, 4), 32, 0, stream>>>(X, ew1, eb1, Hbuf, 1024);

    // enc2: ml = h @ enc_w2 + b2;  M=8192, N=128, K=128
    wmma_gemm_grouped<8, 512, 128, 1, 128 * 128, 128, 128, 512, 128, 1, 0>
        <<<dim3(512, 1, 4), 32, 0, stream>>>(Hbuf, ew2, eb2, mlb, 128);

    // posterior selection + reparameterization -> z (8192,64)
    envae_combine<<<(8192 * 64) / 256, 256, 0, stream>>>(mlb, eps, sidx, z);

    // dec1: hd = relu(z @ dec_w1 + b1);  M=8192, N=64 (NT=4), K=64; z shared across g (LDA_G=0)
    wmma_gemm_grouped<4, 64, 0, 1, 64 * 64, 64, 64, 256, 64, 1, 1>
        <<<dim3(512, 1, 4), 32, 0, stream>>>(z, dw1, db1, hd, 64);

    // dec2: recon[b, g + 4*o] = hd @ dec_w2 + b2;  M=8192, N=1024 (NT=8 -> grid.y=8), K=64
    //   scatter via LDC_COL=4, LDC_G=1 — every output column written exactly once.
    wmma_gemm_grouped<8, 256, 64, 1, 64 * 1024, 1024, 1024, 4096, 1, 4, 0>
        <<<dim3(512, 8, 4), 32, 0, stream>>>(hd, dw2, db2, recon, 64);
}